// MPNN_49632642073098
// MI455X (gfx1250) — compile-verified
//
#include <hip/hip_runtime.h>

typedef __attribute__((ext_vector_type(2))) float v2f;
typedef __attribute__((ext_vector_type(8))) float v8f;

constexpr int NNODES = 20000;
constexpr int NEDGES = 500000;
constexpr int NEF    = 16;     // edge feature dim; node feature dim is also 16 every layer
constexpr float EPSV = 1e-5f;

// ---------------------------------------------------------------------------
// generic zero-fill (graph-capture safe; we own all scratch init)
// ---------------------------------------------------------------------------
__global__ void zero_kernel(float* __restrict__ p, int n) {
  for (int i = blockIdx.x * blockDim.x + threadIdx.x; i < n;
       i += gridDim.x * blockDim.x)
    p[i] = 0.0f;
}

// ---------------------------------------------------------------------------
// Node transform: U[n, f*DOUT+o] = sum_i h[n,i] * W[f, i*DOUT+o]
//                 U[n, NEF*DOUT+o] = sum_i h[n,i] * be[i*DOUT+o]   (c-term)
// One 256-thread block = 8 waves; each wave owns a 16-node tile.
// B (16 x NCT*16, permuted from W/be) staged in LDS once per block,
// *pair-interleaved along k* so each WMMA B operand is one ds_load_b64:
//   Bs[((k>>1)*BCOLS + col)*2 + (k&1)]
// GEMM via chained V_WMMA_F32_16X16X4_F32 (fp32 matrix pipe, K=16 in 4 steps).
// ---------------------------------------------------------------------------
template <int DOUT, int NCT>
__global__ __launch_bounds__(256) void node_transform_kernel(
    const float* __restrict__ h, const float* __restrict__ W,
    const float* __restrict__ be, float* __restrict__ U) {
  constexpr int BCOLS = NCT * 16;
  __shared__ float Bs[16 * BCOLS];

  const int tid = threadIdx.x;
  // cooperative permuted + pair-interleaved load of B into LDS
  for (int idx = tid; idx < 16 * BCOLS; idx += 256) {
    const int i = idx / BCOLS;   // k-row (input channel), 0..15
    const int col = idx % BCOLS;
    float v = 0.0f;
    if (col < NEF * DOUT) {
      const int f = col / DOUT;
      const int o = col % DOUT;
      v = W[f * (16 * DOUT) + i * DOUT + o];
    } else if (col < NEF * DOUT + DOUT) {
      v = be[i * DOUT + (col - NEF * DOUT)];
    }
    Bs[((i >> 1) * BCOLS + col) * 2 + (i & 1)] = v;
  }
  __syncthreads();

  const int wave = tid >> 5;
  const int lane = tid & 31;
  const int tile = blockIdx.x * 8 + wave;
  if (tile * 16 >= NNODES) return;  // wave-uniform: EXEC stays all-ones for WMMA

  // A tile (16x16 fp32, split into four 16x4 WMMA operands)
  // ISA layout: lanes 0-15 hold K,K+1 ; lanes 16-31 hold K+2,K+3 -> one b64/load
  const int row  = lane & 15;
  const int half = lane >> 4;  // 0: k-pair {0,1}, 1: k-pair {2,3} within chunk
  const v2f* __restrict__ hp =
      (const v2f*)(h + (size_t)(tile * 16 + row) * 16);
  v2f a[4];
#pragma unroll
  for (int kk = 0; kk < 4; ++kk) a[kk] = hp[2 * kk + half];

  const v2f* __restrict__ Bp = (const v2f*)Bs;  // pair view of LDS
  const int col = lane & 15;
#pragma unroll
  for (int ct = 0; ct < NCT; ++ct) {
    v8f c = {};
#pragma unroll
    for (int kk = 0; kk < 4; ++kk) {
      // k-pair index = 2*kk + half  ->  single aligned ds_load_b64
      const v2f b = Bp[(2 * kk + half) * BCOLS + ct * 16 + col];
      c = __builtin_amdgcn_wmma_f32_16x16x4_f32(false, a[kk], false, b,
                                                (short)0, c, false, false);
    }
    // D layout: VGPR v, lane L -> M = v + 8*(L>>4), N = L&15
#pragma unroll
    for (int v = 0; v < 8; ++v) {
      const int m = v + 8 * half;
      U[(size_t)(tile * 16 + m) * BCOLS + ct * 16 + col] = c[v];
    }
  }
}

// ---------------------------------------------------------------------------
// Edge kernel: m[e,o] = C[src,o] + sum_f efeat[e,f]*U[src, f*DOUT+o]
// then atomic scatter-add into agg[dst, o].  DOUT lanes per edge; U rows are
// L2-resident (21.8 MB << 192 MB L2); lane index o makes U loads coalesced.
// ---------------------------------------------------------------------------
template <int DOUT, int NCT>
__global__ __launch_bounds__(256) void edge_kernel(
    const float* __restrict__ efeat, const int* __restrict__ src,
    const int* __restrict__ dst, const float* __restrict__ U,
    float* __restrict__ agg) {
  constexpr int BCOLS = NCT * 16;
  const int gid = blockIdx.x * 256 + threadIdx.x;
  const int e = gid / DOUT;
  const int o = gid % DOUT;
  if (e >= NEDGES) return;

  const int s = src[e];
  const int d = dst[e];
  const float* __restrict__ Ur = U + (size_t)s * BCOLS;

  const float4* __restrict__ ef4 =
      (const float4*)(efeat + (size_t)e * NEF);
  const float4 e0 = ef4[0], e1 = ef4[1], e2 = ef4[2], e3 = ef4[3];

  float acc = Ur[NEF * DOUT + o];  // c-term (h[src] @ be)
  acc += e0.x * Ur[0 * DOUT + o];
  acc += e0.y * Ur[1 * DOUT + o];
  acc += e0.z * Ur[2 * DOUT + o];
  acc += e0.w * Ur[3 * DOUT + o];
  acc += e1.x * Ur[4 * DOUT + o];
  acc += e1.y * Ur[5 * DOUT + o];
  acc += e1.z * Ur[6 * DOUT + o];
  acc += e1.w * Ur[7 * DOUT + o];
  acc += e2.x * Ur[8 * DOUT + o];
  acc += e2.y * Ur[9 * DOUT + o];
  acc += e2.z * Ur[10 * DOUT + o];
  acc += e2.w * Ur[11 * DOUT + o];
  acc += e3.x * Ur[12 * DOUT + o];
  acc += e3.y * Ur[13 * DOUT + o];
  acc += e3.z * Ur[14 * DOUT + o];
  acc += e3.w * Ur[15 * DOUT + o];

  atomicAdd(&agg[(size_t)d * DOUT + o], acc);
}

// ---------------------------------------------------------------------------
// y = relu(agg + bias) written in place; accumulate per-channel sum / sumsq
// (LDS tree-reduce, 32 atomics per block). channel == tid&15 is loop-invariant
// because the grid stride is a multiple of 16.
// ---------------------------------------------------------------------------
__global__ __launch_bounds__(256) void relu_bias_stats_kernel(
    float* __restrict__ h, const float* __restrict__ bias,
    float* __restrict__ stats) {
  __shared__ float ssum[256];
  __shared__ float ssq[256];
  const int tid = threadIdx.x;
  const int c = tid & 15;
  const float bv = bias[c];
  float bsum = 0.0f, bsq = 0.0f;
  const int stride = gridDim.x * 256;
  for (int idx = blockIdx.x * 256 + tid; idx < NNODES * 16; idx += stride) {
    float y = h[idx] + bv;
    y = fmaxf(y, 0.0f);
    h[idx] = y;
    bsum += y;
    bsq += y * y;
  }
  ssum[tid] = bsum;
  ssq[tid] = bsq;
  __syncthreads();
  for (int off = 128; off >= 16; off >>= 1) {
    if (tid < off) {
      ssum[tid] += ssum[tid + off];
      ssq[tid] += ssq[tid + off];
    }
    __syncthreads();
  }
  if (tid < 16) {
    atomicAdd(&stats[tid], ssum[tid]);
    atomicAdd(&stats[16 + tid], ssq[tid]);
  }
}

__global__ void bn_finalize_kernel(float* __restrict__ stats) {
  const int t = threadIdx.x;
  if (t < 16) {
    const float inv = 1.0f / (float)NNODES;
    const float mu = stats[t] * inv;
    const float var = stats[16 + t] * inv - mu * mu;  // biased var
    stats[32 + t] = mu;
    stats[48 + t] = rsqrtf(var + EPSV);
  }
}

__global__ __launch_bounds__(256) void bn_apply_kernel(
    float* __restrict__ h, const float* __restrict__ stats,
    const float* __restrict__ g, const float* __restrict__ beta) {
  const int idx = blockIdx.x * 256 + threadIdx.x;
  if (idx >= NNODES * 16) return;
  const int c = idx & 15;
  h[idx] = g[c] * (h[idx] - stats[32 + c]) * stats[48 + c] + beta[c];
}

__global__ __launch_bounds__(256) void log_softmax_kernel(
    const float* __restrict__ agg, const float* __restrict__ bias,
    float* __restrict__ out) {
  const int n = blockIdx.x * 256 + threadIdx.x;
  if (n >= NNODES) return;
  float z[8];
  float mx = -3.0e38f;
#pragma unroll
  for (int o = 0; o < 8; ++o) {
    z[o] = agg[(size_t)n * 8 + o] + bias[o];
    mx = fmaxf(mx, z[o]);
  }
  float se = 0.0f;
#pragma unroll
  for (int o = 0; o < 8; ++o) se += expf(z[o] - mx);
  const float lse = mx + logf(se);
#pragma unroll
  for (int o = 0; o < 8; ++o) out[(size_t)n * 8 + o] = z[o] - lse;
}

// ---------------------------------------------------------------------------
extern "C" void kernel_launch(void* const* d_in, const int* in_sizes, int n_in,
                              void* d_out, int out_size, void* d_ws,
                              size_t ws_size, hipStream_t stream) {
  const float* x     = (const float*)d_in[0];
  const float* efeat = (const float*)d_in[1];
  const int*   src   = (const int*)d_in[2];
  const int*   dst   = (const int*)d_in[3];
  const float* W0    = (const float*)d_in[4];
  const float* b0    = (const float*)d_in[5];
  const float* bias0 = (const float*)d_in[6];
  const float* g0    = (const float*)d_in[7];
  const float* beta0 = (const float*)d_in[8];
  const float* W1    = (const float*)d_in[9];
  const float* b1    = (const float*)d_in[10];
  const float* bias1 = (const float*)d_in[11];
  const float* g1    = (const float*)d_in[12];
  const float* beta1 = (const float*)d_in[13];
  const float* W2    = (const float*)d_in[14];
  const float* b2    = (const float*)d_in[15];
  const float* bias2 = (const float*)d_in[16];
  float* out = (float*)d_out;

  // workspace layout (floats): hA[N*16] | hB[N*16] | stats[64] | U[N*272]
  float* ws    = (float*)d_ws;
  float* hA    = ws;
  float* hB    = hA + NNODES * 16;
  float* stats = hB + NNODES * 16;
  float* U     = stats + 64;

  const int NT_BLOCKS = (NNODES / 16 + 7) / 8;  // 157 blocks of 8 waves

  // ---------------- layer 0: NNConv(16->16) + relu + BN ----------------
  zero_kernel<<<256, 256, 0, stream>>>(hA, NNODES * 16);
  zero_kernel<<<1, 64, 0, stream>>>(stats, 64);
  node_transform_kernel<16, 17><<<NT_BLOCKS, 256, 0, stream>>>(x, W0, b0, U);
  edge_kernel<16, 17><<<(NEDGES * 16) / 256, 256, 0, stream>>>(efeat, src, dst,
                                                               U, hA);
  relu_bias_stats_kernel<<<160, 256, 0, stream>>>(hA, bias0, stats);
  bn_finalize_kernel<<<1, 32, 0, stream>>>(stats);
  bn_apply_kernel<<<(NNODES * 16) / 256, 256, 0, stream>>>(hA, stats, g0,
                                                           beta0);

  // ---------------- layer 1: NNConv(16->16) + relu + BN ----------------
  zero_kernel<<<256, 256, 0, stream>>>(hB, NNODES * 16);
  zero_kernel<<<1, 64, 0, stream>>>(stats, 64);
  node_transform_kernel<16, 17><<<NT_BLOCKS, 256, 0, stream>>>(hA, W1, b1, U);
  edge_kernel<16, 17><<<(NEDGES * 16) / 256, 256, 0, stream>>>(efeat, src, dst,
                                                               U, hB);
  relu_bias_stats_kernel<<<160, 256, 0, stream>>>(hB, bias1, stats);
  bn_finalize_kernel<<<1, 32, 0, stream>>>(stats);
  bn_apply_kernel<<<(NNODES * 16) / 256, 256, 0, stream>>>(hB, stats, g1,
                                                           beta1);

  // ---------------- layer 2: NNConv(16->8) + log_softmax ----------------
  zero_kernel<<<64, 256, 0, stream>>>(hA, NNODES * 8);  // reuse hA as agg2
  node_transform_kernel<8, 9><<<NT_BLOCKS, 256, 0, stream>>>(hB, W2, b2, U);
  edge_kernel<8, 9><<<(NEDGES * 8) / 256, 256, 0, stream>>>(efeat, src, dst, U,
                                                            hA);
  log_softmax_kernel<<<(NNODES + 255) / 256, 256, 0, stream>>>(hA, bias2, out);
}